// SSM_65137474011928
// MI455X (gfx1250) — compile-verified
//
#include <hip/hip_runtime.h>
#include <math.h>

typedef __attribute__((ext_vector_type(2))) float v2f;
typedef __attribute__((ext_vector_type(4))) float v4f;
typedef __attribute__((ext_vector_type(8))) float v8f;

#define SSM_BATCH 8
#define SSM_SEQ   2048
#define SSM_D     64
#define SSM_N     64
#define SSM_M     (SSM_BATCH * SSM_SEQ)   // 16384 rows
#define SSM_NCOL  192                      // 64 B-cols | 64 dt-cols | 64 C-cols

// ---------------------------------------------------------------------------
// Kernel 1: fused projection GEMM with V_WMMA_F32_16X16X4_F32.
//   out[m][j] = sum_k x[m][k] * Wbig[k][j],  M=16384, N=192, K=64
//   Wbig cols 0..63   = Wx[B rows]       -> Bm
//   Wbig cols 64..127 = Wx[dt rows]@Wdt^T -> dt (+bias, clip +-0.1)
//   Wbig cols 128..191= Wx[C rows]       -> Cm
// Weights are built per-block directly into LDS, pre-swizzled into the
// per-lane WMMA B-matrix layout (lane 0-15: K=k0,k0+1; lane 16-31: K=k0+2,k0+3).
// ---------------------------------------------------------------------------
__global__ __launch_bounds__(256)
void ssm_proj_kernel(const float* __restrict__ x, const float* __restrict__ Wx,
                     const float* __restrict__ Wdt, const float* __restrict__ bdt,
                     float* __restrict__ Bm, float* __restrict__ dtc,
                     float* __restrict__ Cm)
{
    __shared__ v2f wlds[12 * 16 * 32];   // 48 KB: [j-tile][k-step][lane]
    const int tid = threadIdx.x;

    // Build swizzled weights (one-time per block; weights are tiny).
    for (int idx = tid; idx < 12 * 16 * 32; idx += 256) {
        const int j    = idx >> 9;          // /512 -> n-tile
        const int kk   = (idx >> 5) & 15;   // k-step (K advances by 4)
        const int lane = idx & 31;
        const int col  = j * 16 + (lane & 15);
        const int k0   = kk * 4 + ((lane >> 4) << 1);
        float w0, w1;
        if (col < 64) {                     // B projection rows of Wx
            w0 = Wx[col * 64 + k0];
            w1 = Wx[col * 64 + k0 + 1];
        } else if (col < 128) {             // dt: fold Wx[64+r][:] through Wdt
            const int e = col - 64;
            float a0 = 0.f, a1 = 0.f;
            #pragma unroll
            for (int r = 0; r < 4; ++r) {
                const float wd = Wdt[e * 4 + r];
                a0 += Wx[(64 + r) * 64 + k0]     * wd;
                a1 += Wx[(64 + r) * 64 + k0 + 1] * wd;
            }
            w0 = a0; w1 = a1;
        } else {                            // C projection rows of Wx
            const int n = col - 128;
            w0 = Wx[(68 + n) * 64 + k0];
            w1 = Wx[(68 + n) * 64 + k0 + 1];
        }
        v2f w; w.x = w0; w.y = w1;
        wlds[idx] = w;
    }
    __syncthreads();

    const int lane = tid & 31;
    const int wave = tid >> 5;
    const int m0   = blockIdx.x * 128 + wave * 16;   // 8 waves x 16-row tiles
    const int row  = m0 + (lane & 15);
    const int kb   = (lane >> 4) << 1;

    // A tile (16x64 fp32) in registers: 16 k-steps x float2 per lane.
    v2f a[16];
    #pragma unroll
    for (int kk = 0; kk < 16; ++kk) {
        const float* p = x + (size_t)row * 64 + kk * 4 + kb;
        v2f t; t.x = p[0]; t.y = p[1];
        a[kk] = t;
    }

    const int colB  = (lane & 15);
    const int mBase = m0 + ((lane >> 4) << 3);   // lanes 16-31 hold rows M=8..15

    #pragma unroll
    for (int j = 0; j < 12; ++j) {
        v8f acc = {};
        #pragma unroll
        for (int kk = 0; kk < 16; ++kk) {
            const v2f bm = wlds[(j * 16 + kk) * 32 + lane];
            acc = __builtin_amdgcn_wmma_f32_16x16x4_f32(
                false, a[kk], false, bm, (short)0, acc, false, false);
        }
        const int col = j * 16 + colB;
        if (j < 4) {
            #pragma unroll
            for (int r = 0; r < 8; ++r)
                Bm[(size_t)(mBase + r) * 64 + col] = acc[r];
        } else if (j < 8) {
            const int e  = col - 64;
            const float bb = bdt[e];
            #pragma unroll
            for (int r = 0; r < 8; ++r) {
                float v = acc[r] + bb;
                v = fminf(fmaxf(v, -0.1f), 0.1f);
                dtc[(size_t)(mBase + r) * 64 + e] = v;
            }
        } else {
            const int n = col - 128;
            #pragma unroll
            for (int r = 0; r < 8; ++r)
                Cm[(size_t)(mBase + r) * 64 + n] = acc[r];
        }
    }
}

// ---------------------------------------------------------------------------
// Kernel 2: per-(b,t) scalar coefficients.
//   lam = trace(deltaA) = sum_i A[i][i]*dt[i],  A = -exp(A_log)
//   phi = expm1(lam)/lam        (-> expm(deltaA) = I + phi*deltaA, rank-1)
//   psi = expm1(lam)/(lam+1e-5) (Sherman-Morrison closed form of pinv chain)
// ---------------------------------------------------------------------------
__global__ __launch_bounds__(256)
void ssm_coef_kernel(const float* __restrict__ dtc, const float* __restrict__ A_log,
                     float* __restrict__ phi, float* __restrict__ psi)
{
    const int m = blockIdx.x * 256 + threadIdx.x;   // 64 blocks -> 16384 rows
    float lam = 0.f;
    #pragma unroll 8
    for (int e = 0; e < 64; ++e) {
        const float adiag = -expf(A_log[e * 64 + e]);
        lam += adiag * dtc[(size_t)m * 64 + e];
    }
    const float em = expm1f(lam);
    phi[m] = (lam == 0.0f) ? 1.0f : em / lam;
    psi[m] = em / (lam + 1e-5f);
}

// ---------------------------------------------------------------------------
// Kernel 3: sequential scan over t (elementwise recurrence per (d,n)).
// Grid: 8 batches x 8 n-chunks = 64 blocks; 128 threads; thread owns a
// float4 of state h[d][n0..n0+3]. Streams 256 MiB of h with NT b128 stores;
// next-step inputs are register-pipelined + global_prefetch warms caches.
// ---------------------------------------------------------------------------
__global__ __launch_bounds__(128)
void ssm_scan_kernel(const float* __restrict__ x, const float* __restrict__ A_log,
                     const float* __restrict__ Bm, const float* __restrict__ dtc,
                     const float* __restrict__ phi, const float* __restrict__ psi,
                     float* __restrict__ hout)
{
    const int b     = blockIdx.x >> 3;
    const int nbase = (blockIdx.x & 7) << 3;
    const int tid   = threadIdx.x;
    const int d     = tid >> 1;                   // 0..63
    const int n0    = nbase + ((tid & 1) << 2);   // float4-aligned column

    v4f av = *reinterpret_cast<const v4f*>(A_log + d * 64 + n0);
    av.x = -expf(av.x); av.y = -expf(av.y); av.z = -expf(av.z); av.w = -expf(av.w);

    v4f diag;
    diag.x = (d == n0 + 0) ? 1.f : 0.f;
    diag.y = (d == n0 + 1) ? 1.f : 0.f;
    diag.z = (d == n0 + 2) ? 1.f : 0.f;
    diag.w = (d == n0 + 3) ? 1.f : 0.f;

    const size_t mb = (size_t)b * SSM_SEQ;

    // Prime the software pipeline with t=0 inputs.
    float dtv = dtc[mb * 64 + d];
    float xv  = x  [mb * 64 + d];
    float ph  = phi[mb];
    float ps  = psi[mb];
    v4f   Bv  = *reinterpret_cast<const v4f*>(Bm + mb * 64 + n0);
    v4f   h; h.x = h.y = h.z = h.w = 0.f;

    for (int t = 0; t < SSM_SEQ; ++t) {
        const size_t m  = mb + t;
        const size_t mn = mb + ((t + 1 < SSM_SEQ) ? (t + 1) : t);

        if (t + 16 < SSM_SEQ) {             // warm caches well ahead
            const size_t mf = m + 16;
            __builtin_prefetch(dtc + mf * 64 + d, 0, 0);
            __builtin_prefetch(x   + mf * 64 + d, 0, 0);
            __builtin_prefetch(Bm  + mf * 64 + n0, 0, 0);
            __builtin_prefetch(phi + mf, 0, 0);
            __builtin_prefetch(psi + mf, 0, 0);
        }

        // Issue next-step loads before consuming current values.
        const float dtn = dtc[mn * 64 + d];
        const float xn  = x  [mn * 64 + d];
        const float phn = phi[mn];
        const float psn = psi[mn];
        const v4f   Bn  = *reinterpret_cast<const v4f*>(Bm + mn * 64 + n0);

        const float ca = ph * dtv;          // phi * dt[d]
        const float cb = ps * dtv * xv;     // psi * dt[d] * x[d]
        h.x = fminf(fmaxf(fmaf(h.x, diag.x + ca * av.x, cb * Bv.x), -1e6f), 1e6f);
        h.y = fminf(fmaxf(fmaf(h.y, diag.y + ca * av.y, cb * Bv.y), -1e6f), 1e6f);
        h.z = fminf(fmaxf(fmaf(h.z, diag.z + ca * av.z, cb * Bv.z), -1e6f), 1e6f);
        h.w = fminf(fmaxf(fmaf(h.w, diag.w + ca * av.w, cb * Bv.w), -1e6f), 1e6f);

        __builtin_nontemporal_store(
            h, reinterpret_cast<v4f*>(hout + m * 4096 + (size_t)d * 64 + n0));

        dtv = dtn; xv = xn; ph = phn; ps = psn; Bv = Bn;
    }
}

// ---------------------------------------------------------------------------
// Kernel 4: final-step output  y[b][d] = h_last[d]. C_last + D[d]*x_last[d]
// (reads h already written into d_out by the scan; fully deterministic).
// ---------------------------------------------------------------------------
__global__ __launch_bounds__(64)
void ssm_y_kernel(const float* __restrict__ x, const float* __restrict__ Dv,
                  const float* __restrict__ Cm, const float* __restrict__ hst,
                  float* __restrict__ yout)
{
    const int b = blockIdx.x;
    const int d = threadIdx.x;
    const size_t m = (size_t)b * SSM_SEQ + (SSM_SEQ - 1);
    const float* hrow = hst + m * 4096 + (size_t)d * 64;
    const float* crow = Cm  + m * 64;
    float acc = 0.f;
    #pragma unroll
    for (int n = 0; n < 64; n += 4) {
        v4f hv = *reinterpret_cast<const v4f*>(hrow + n);
        v4f cv = *reinterpret_cast<const v4f*>(crow + n);
        acc += hv.x * cv.x + hv.y * cv.y + hv.z * cv.z + hv.w * cv.w;
    }
    yout[b * 64 + d] = acc + Dv[d] * x[m * 64 + d];
}

// ---------------------------------------------------------------------------
extern "C" void kernel_launch(void* const* d_in, const int* in_sizes, int n_in,
                              void* d_out, int out_size, void* d_ws, size_t ws_size,
                              hipStream_t stream)
{
    const float* x     = (const float*)d_in[0];
    const float* Wx    = (const float*)d_in[1];
    const float* Wdt   = (const float*)d_in[2];
    const float* bdt   = (const float*)d_in[3];
    const float* A_log = (const float*)d_in[4];
    const float* Dv    = (const float*)d_in[5];

    float* out  = (float*)d_out;
    float* hout = out + SSM_BATCH * SSM_D;   // hs after ys[-1] (512 floats)

    char* ws = (char*)d_ws;
    float* Bm  = (float*)(ws);                                   // 4 MiB
    float* dtc = (float*)(ws + (size_t)4  * 1024 * 1024);        // 4 MiB
    float* Cm  = (float*)(ws + (size_t)8  * 1024 * 1024);        // 4 MiB
    float* phi = (float*)(ws + (size_t)12 * 1024 * 1024);        // 64 KiB
    float* psi = (float*)(ws + (size_t)12 * 1024 * 1024 + 64 * 1024);

    ssm_proj_kernel<<<SSM_M / 128, 256, 0, stream>>>(x, Wx, Wdt, bdt, Bm, dtc, Cm);
    ssm_coef_kernel<<<SSM_M / 256, 256, 0, stream>>>(dtc, A_log, phi, psi);
    ssm_scan_kernel<<<SSM_BATCH * 8, 128, 0, stream>>>(x, A_log, Bm, dtc, phi, psi, hout);
    ssm_y_kernel<<<SSM_BATCH, 64, 0, stream>>>(x, Dv, Cm, hout, out);
}